// GCNEncoder_32512902431448
// MI455X (gfx1250) — compile-verified
//
#include <hip/hip_runtime.h>
#include <stdint.h>

// ---------------------------------------------------------------------------
// GCN on fully-connected graph, N=1024, IN=64, HID=128, OUT=64, gfx1250.
// Aggregation materialized as dense normalized adjacency Ahat (4MB, L2-resident),
// executed as FP32 WMMA GEMMs (V_WMMA_F32_16X16X4_F32). The dominant K=1024
// GEMMs stage the shared B panel in LDS via GLOBAL_LOAD_ASYNC_TO_LDS_B128
// (ASYNCcnt) with double buffering.
// ---------------------------------------------------------------------------

typedef __attribute__((ext_vector_type(2))) float v2f;
typedef __attribute__((ext_vector_type(8))) float v8f;

#define NN   1024   // num nodes
#define IND  64     // input dim
#define HIDD 128    // hidden dim
#define OUTD 64     // output dim
#define KC   64     // K-chunk staged in LDS for the big GEMMs

// --- init: zero Ahat, deg = 1.0 (self-loop contribution) -------------------
__global__ __launch_bounds__(256) void k_init(float4* __restrict__ A4,
                                              float* __restrict__ deg) {
  int i = blockIdx.x * blockDim.x + threadIdx.x;
  if (i < (NN * NN) / 4) {
    float4 z; z.x = 0.f; z.y = 0.f; z.z = 0.f; z.w = 0.f;
    A4[i] = z;
  }
  if (i < NN) deg[i] = 1.0f;
}

// --- degree scatter --------------------------------------------------------
__global__ __launch_bounds__(256) void k_deg(const int* __restrict__ dst,
                                             float* __restrict__ deg, int E) {
  int e = blockIdx.x * blockDim.x + threadIdx.x;
  if (e < E) atomicAdd(&deg[dst[e]], 1.0f);
}

// --- dinv = rsqrt(deg) -----------------------------------------------------
__global__ __launch_bounds__(256) void k_dinv(const float* __restrict__ deg,
                                              float* __restrict__ dinv) {
  int i = blockIdx.x * blockDim.x + threadIdx.x;
  if (i < NN) dinv[i] = rsqrtf(deg[i]);
}

// --- build Ahat[dst][src] += dinv[src]*dinv[dst]; diag += dinv[i]^2 --------
__global__ __launch_bounds__(256) void k_buildA(const int* __restrict__ src,
                                                const int* __restrict__ dst,
                                                const float* __restrict__ dinv,
                                                float* __restrict__ Amat, int E) {
  int e = blockIdx.x * blockDim.x + threadIdx.x;
  if (e < E) {
    int s = src[e], d = dst[e];
    atomicAdd(&Amat[(size_t)d * NN + s], dinv[s] * dinv[d]);
  } else if (e < E + NN) {
    int i = e - E;
    atomicAdd(&Amat[(size_t)i * NN + i], dinv[i] * dinv[i]);
  }
}

// --- async global -> LDS helpers (CDNA5, ASYNCcnt-tracked) -----------------
__device__ __forceinline__ void async_b128_to_lds(const float* gptr, void* lptr) {
  uint32_t lds_addr = (uint32_t)(uintptr_t)lptr;   // addr[31:0] = LDS byte offset
  uint64_t gaddr = (uint64_t)(uintptr_t)gptr;
  asm volatile("global_load_async_to_lds_b128 %0, %1, off"
               :: "v"(lds_addr), "v"(gaddr)
               : "memory");
}
__device__ __forceinline__ void wait_async_le1() {
  asm volatile("s_wait_asynccnt 0x1" ::: "memory");
}
__device__ __forceinline__ void wait_async_le0() {
  asm volatile("s_wait_asynccnt 0x0" ::: "memory");
}

// --- simple FP32 WMMA GEMM (small K, tiny B): one 16x16 tile per wave ------
// C[M x Nc] = A[M x K] @ B[K x Nc], row-major. Optional relu(C + bias[col]).
__global__ __launch_bounds__(256)
void k_gemm_wmma_f32(float* __restrict__ C, const float* __restrict__ A,
                     const float* __restrict__ B, const float* __restrict__ bias,
                     int M, int Nc, int K, int relu) {
  const int lane = threadIdx.x & 31;
  const int wave = (int)((blockIdx.x * blockDim.x + threadIdx.x) >> 5);
  const int ntn  = Nc >> 4;
  const int total = (M >> 4) * ntn;
  if (wave >= total) return;                // wave-uniform: EXEC all-1 inside

  const int tm = wave / ntn;
  const int tn = wave - tm * ntn;
  const int mn = lane & 15;
  const int kb = (lane >> 4) << 1;          // 0 or 2

  const float* __restrict__ arow = A + (size_t)(tm * 16 + mn) * K + kb;
  const float* __restrict__ bcol = B + (size_t)kb * Nc + (tn * 16 + mn);

  v8f acc = {0.f, 0.f, 0.f, 0.f, 0.f, 0.f, 0.f, 0.f};
  for (int k = 0; k < K; k += 4) {
    v2f a = *(const v2f*)(arow + k);
    v2f b;
    b.x = bcol[(size_t)k * Nc];
    b.y = bcol[(size_t)(k + 1) * Nc];
    acc = __builtin_amdgcn_wmma_f32_16x16x4_f32(
        false, a, false, b, (short)0, acc, false, false);
  }

  const int col = tn * 16 + mn;
  const float bv = bias ? bias[col] : 0.0f;
  const int rbase = tm * 16 + ((lane >> 4) << 3);
#pragma unroll
  for (int r = 0; r < 8; ++r) {
    float v = acc[r] + bv;
    if (relu) v = fmaxf(v, 0.0f);
    C[(size_t)(rbase + r) * Nc + col] = v;
  }
}

// --- LDS-staged FP32 WMMA GEMM for the big K=1024 aggregation GEMMs --------
// Block (8 waves) computes a 128x16 output tile: wave w -> rows [w*16,+16).
// B panel (K x 16) staged chunk-wise (KC x 16) in double-buffered LDS via
// global_load_async_to_lds_b128; all 8 waves share the staged panel.
// Requires: M % 128 == 0, Nc % 16 == 0, K % KC == 0.
__global__ __launch_bounds__(256)
void k_gemm_wmma_lds(float* __restrict__ C, const float* __restrict__ A,
                     const float* __restrict__ B, const float* __restrict__ bias,
                     int Nc, int K, int relu) {
  __shared__ float bsh[2][KC * 16];

  const int tid  = threadIdx.x;
  const int lane = tid & 31;
  const int wave = tid >> 5;                // 0..7
  const int n0   = blockIdx.x * 16;         // output col base
  const int m0   = blockIdx.y * 128 + wave * 16;

  const int mn = lane & 15;
  const int kb = (lane >> 4) << 1;          // 0 or 2

  // copy role: thread -> (row r of chunk, 16B slice c4)
  const int r  = tid >> 2;                  // 0..63
  const int c4 = (tid & 3) << 2;            // 0,4,8,12 floats

  const float* __restrict__ arow = A + (size_t)(m0 + mn) * K + kb;
  const int nch = K / KC;

  // prologue: stage chunk 0 into buffer 0
  async_b128_to_lds(B + (size_t)r * Nc + n0 + c4, &bsh[0][r * 16 + c4]);

  v8f acc = {0.f, 0.f, 0.f, 0.f, 0.f, 0.f, 0.f, 0.f};

  for (int ch = 0; ch < nch; ++ch) {
    if (ch + 1 < nch) {   // prefetch next chunk into the other buffer
      async_b128_to_lds(B + (size_t)((ch + 1) * KC + r) * Nc + n0 + c4,
                        &bsh[(ch + 1) & 1][r * 16 + c4]);
      wait_async_le1();   // in-order: oldest (chunk ch) is complete
    } else {
      wait_async_le0();
    }
    __syncthreads();      // all waves' pieces of chunk ch visible

    const float* __restrict__ bb = &bsh[ch & 1][0];
    const int k0 = ch * KC;
#pragma unroll
    for (int kk = 0; kk < KC; kk += 4) {
      v2f a = *(const v2f*)(arow + k0 + kk);
      v2f b;
      b.x = bb[(kk + kb) * 16 + mn];
      b.y = bb[(kk + kb + 1) * 16 + mn];
      acc = __builtin_amdgcn_wmma_f32_16x16x4_f32(
          false, a, false, b, (short)0, acc, false, false);
    }
    __syncthreads();      // done reading this buffer before it is rewritten
  }

  const int col = n0 + mn;
  const float bv = bias ? bias[col] : 0.0f;
  const int rbase = m0 + ((lane >> 4) << 3);
#pragma unroll
  for (int rr = 0; rr < 8; ++rr) {
    float v = acc[rr] + bv;
    if (relu) v = fmaxf(v, 0.0f);
    C[(size_t)(rbase + rr) * Nc + col] = v;
  }
}

// --- row mean over OUTD columns -------------------------------------------
__global__ __launch_bounds__(256) void k_rowmean(const float* __restrict__ H,
                                                 float* __restrict__ out) {
  int i = blockIdx.x * blockDim.x + threadIdx.x;
  if (i < NN) {
    float s = 0.f;
#pragma unroll
    for (int j = 0; j < OUTD; ++j) s += H[(size_t)i * OUTD + j];
    out[i] = s * (1.0f / OUTD);
  }
}

extern "C" void kernel_launch(void* const* d_in, const int* in_sizes, int n_in,
                              void* d_out, int out_size, void* d_ws, size_t ws_size,
                              hipStream_t stream) {
  const float* x  = (const float*)d_in[0];   // [1024 x 64]
  const float* W1 = (const float*)d_in[1];   // [64 x 128]
  const float* b1 = (const float*)d_in[2];   // [128]
  const float* W2 = (const float*)d_in[3];   // [128 x 64]
  const float* b2 = (const float*)d_in[4];   // [64]
  const int*  src = (const int*)d_in[5];     // [E]
  const int*  dst = (const int*)d_in[6];     // [E]
  const int E = in_sizes[5];

  // Workspace layout (~5.03 MB total):
  char* ws = (char*)d_ws;
  float* Amat = (float*)ws;                                   // 1024*1024 f32 (4MB)
  float* deg  = (float*)(ws + (size_t)NN * NN * sizeof(float));
  float* dinv = deg + NN;
  float* buf1 = dinv + NN;                                    // up to 1024*128 f32
  float* buf2 = buf1 + (size_t)NN * HIDD;                     // up to 1024*128 f32

  const int TB = 256;

  // 1) init Ahat=0, deg=1
  k_init<<<(NN * NN / 4 + TB - 1) / TB, TB, 0, stream>>>((float4*)Amat, deg);
  // 2) degree scatter
  k_deg<<<(E + TB - 1) / TB, TB, 0, stream>>>(dst, deg, E);
  // 3) dinv
  k_dinv<<<(NN + TB - 1) / TB, TB, 0, stream>>>(deg, dinv);
  // 4) build Ahat
  k_buildA<<<(E + NN + TB - 1) / TB, TB, 0, stream>>>(src, dst, dinv, Amat, E);

  // 5) H1pre = X @ W1            [1024x64]@[64x128] -> buf1   (small K)
  {
    int tiles = (NN / 16) * (HIDD / 16);
    k_gemm_wmma_f32<<<(tiles + 7) / 8, TB, 0, stream>>>(
        buf1, x, W1, nullptr, NN, HIDD, IND, 0);
  }
  // 6) H1 = relu(Ahat @ H1pre + b1)   [1024x1024]@[1024x128] -> buf2  (LDS-staged)
  {
    dim3 grid(HIDD / 16, NN / 128);
    k_gemm_wmma_lds<<<grid, TB, 0, stream>>>(buf2, Amat, buf1, b1, HIDD, NN, 1);
  }
  // 7) H2pre = H1 @ W2           [1024x128]@[128x64] -> buf1  (small K)
  {
    int tiles = (NN / 16) * (OUTD / 16);
    k_gemm_wmma_f32<<<(tiles + 7) / 8, TB, 0, stream>>>(
        buf1, buf2, W2, nullptr, NN, OUTD, HIDD, 0);
  }
  // 8) H2 = relu(Ahat @ H2pre + b2)   [1024x1024]@[1024x64] -> buf2  (LDS-staged)
  {
    dim3 grid(OUTD / 16, NN / 128);
    k_gemm_wmma_lds<<<grid, TB, 0, stream>>>(buf2, Amat, buf1, b2, OUTD, NN, 1);
  }
  // 9) out = mean over columns
  k_rowmean<<<(NN + TB - 1) / TB, TB, 0, stream>>>(buf2, (float*)d_out);
}